// GINERegressor_60601988547139
// MI455X (gfx1250) — compile-verified
//
#include <hip/hip_runtime.h>
#include <hip/hip_bf16.h>

#define N_NODES 50000
#define N_EDGES 300000
#define NODE_DIM 11
#define HID 256
#define NLAYERS 4
#define NGRAPH 1024
#define BN_EPS 1e-5f

typedef __attribute__((ext_vector_type(2))) float v2f;
typedef __attribute__((ext_vector_type(8))) float v8f;

// ---------------------------------------------------------------------------
// Node encoder: h = relu(x @ enc_W + enc_b).  K=11 (tiny) -> scalar kernel.
// One block per node, 256 threads = 256 output channels.
// ---------------------------------------------------------------------------
__global__ void encoder_kernel(const float* __restrict__ x,
                               const float* __restrict__ W,   // [11,256]
                               const float* __restrict__ b,   // [256]
                               float* __restrict__ h) {
    __shared__ float xs[NODE_DIM];
    int node = blockIdx.x;
    int c = threadIdx.x;
    if (threadIdx.x < NODE_DIM)
        xs[threadIdx.x] = x[(size_t)node * NODE_DIM + threadIdx.x];
    __syncthreads();
    float acc = b[c];
#pragma unroll
    for (int k = 0; k < NODE_DIM; ++k)
        acc += xs[k] * W[k * HID + c];
    h[(size_t)node * HID + c] = acc > 0.f ? acc : 0.f;
}

// ---------------------------------------------------------------------------
// Fused edge kernel (per layer):
//   e   = edge_attr @ edge_W[l] + edge_b[l]       (K=5)
//   msg = relu(h[src] + e)
//   atomicAdd(aggr[dst], msg)                     -> global_atomic_add_f32
// One wave32 per edge, 8 channels per lane. aggr pre-initialized with h so
// the result is z = h + sum(msg).
// ---------------------------------------------------------------------------
__global__ void edge_aggr_kernel(const float* __restrict__ ea,   // [E,5]
                                 const int*   __restrict__ ei,   // [2,E]
                                 const float* __restrict__ h,    // [N,256]
                                 const float* __restrict__ W,    // [5,256]
                                 const float* __restrict__ b,    // [256]
                                 float* __restrict__ aggr,       // [N,256]
                                 int E) {
    int wid  = (int)((blockIdx.x * blockDim.x + threadIdx.x) >> 5);
    int lane = threadIdx.x & 31;
    if (wid >= E) return;
    int src = ei[wid];
    int dst = ei[E + wid];
    float a0 = ea[(size_t)wid * 5 + 0];
    float a1 = ea[(size_t)wid * 5 + 1];
    float a2 = ea[(size_t)wid * 5 + 2];
    float a3 = ea[(size_t)wid * 5 + 3];
    float a4 = ea[(size_t)wid * 5 + 4];
    const float* hs = h    + (size_t)src * HID;
    float*       ag = aggr + (size_t)dst * HID;
    int c0 = lane * 8;
#pragma unroll
    for (int j = 0; j < 8; ++j) {
        int c = c0 + j;
        float e = b[c] + a0 * W[c] + a1 * W[HID + c] + a2 * W[2 * HID + c]
                       + a3 * W[3 * HID + c] + a4 * W[4 * HID + c];
        float m = hs[c] + e;
        m = m > 0.f ? m : 0.f;
        atomicAdd(&ag[c], m);
    }
}

// ---------------------------------------------------------------------------
// LDS-staged fp32 WMMA GEMM: C = [relu](A[M,K] @ B[K,NC] + bias[NC])
//   grid  = M/16 blocks, 256 threads (8 wave32) per block
//   block stages its 16xK A-tile in LDS once (16*K*4 <= 48KB <= 320KB/WGP)
//   waves split the N dimension: TPW = NC/16/8 column tiles per wave
//   K-loop unrolled by 8 -> 2 WMMAs per tile per iteration, 4 B-loads in
//   flight; A fragments come from LDS (ds_load_b64, 16 rows across lanes ->
//   conflict-free), B streams from L2 (weights are L2-resident).
// Fragment layouts per CDNA5 ISA 7.12.2 (V_WMMA_F32_16X16X4_F32):
//   A 16x4 : lane&15 = row, {v0,v1} = K = 2*hi, 2*hi+1   (hi = lane>>4)
//   B 4x16 : lane&15 = col, {v0,v1} = K = 2*hi, 2*hi+1
//   C      : vgpr i -> row i + 8*hi, col = lane&15
// No divergent branches anywhere -> EXEC all-ones as WMMA requires.
// ---------------------------------------------------------------------------
template <int K, int NC>
__global__ void gemm_wmma_tiled(const float* __restrict__ A,
                                const float* __restrict__ B,
                                const float* __restrict__ bias,
                                float* __restrict__ C,
                                int doRelu) {
    constexpr int TILES_N = NC / 16;
    constexpr int TPW = (TILES_N + 7) / 8;       // column tiles per wave (1|2)
    extern __shared__ float As[];                // [16 * K]

    const int tid  = threadIdx.x;
    const int wave = tid >> 5;
    const int lane = tid & 31;
    const int m0   = blockIdx.x * 16;

    // stage A tile (16 x K) into LDS, float4-vectorized, coalesced
    {
        const float4* Ag  = (const float4*)(A + (size_t)m0 * K);
        float4*       As4 = (float4*)As;
        constexpr int TOT4 = 16 * K / 4;
        for (int i = tid; i < TOT4; i += 256)
            As4[i] = Ag[i];
    }
    __syncthreads();

    const int r  = lane & 15;
    const int hi = lane >> 4;

    v8f acc[TPW] = {};
    const float* Bp[TPW];
#pragma unroll
    for (int t = 0; t < TPW; ++t) {
        int n0 = (wave * TPW + t) * 16;
        Bp[t] = B + (size_t)(2 * hi) * NC + n0 + r;
    }
    const float* Ap = As + (size_t)r * K + 2 * hi;

    for (int k = 0; k < K; k += 8) {
        v2f a0, a1;                              // ds_load_b64 x2
        a0.x = Ap[k];     a0.y = Ap[k + 1];
        a1.x = Ap[k + 4]; a1.y = Ap[k + 5];
#pragma unroll
        for (int t = 0; t < TPW; ++t) {
            if (k + 32 < K)
                __builtin_prefetch(Bp[t] + (size_t)(k + 32) * NC, 0, 1);
            v2f b0, b1;
            b0.x = Bp[t][(size_t)k * NC];
            b0.y = Bp[t][(size_t)(k + 1) * NC];
            b1.x = Bp[t][(size_t)(k + 4) * NC];
            b1.y = Bp[t][(size_t)(k + 5) * NC];
            acc[t] = __builtin_amdgcn_wmma_f32_16x16x4_f32(
                false, a0, false, b0, (short)0, acc[t], false, false);
            acc[t] = __builtin_amdgcn_wmma_f32_16x16x4_f32(
                false, a1, false, b1, (short)0, acc[t], false, false);
        }
    }

#pragma unroll
    for (int t = 0; t < TPW; ++t) {
        int n0 = (wave * TPW + t) * 16;
        float bs = bias ? bias[n0 + r] : 0.f;
        float* Cp = C + (size_t)(m0 + 8 * hi) * NC + n0 + r;
#pragma unroll
        for (int i = 0; i < 8; ++i) {
            float v = acc[t][i] + bs;
            if (doRelu) v = v > 0.f ? v : 0.f;
            Cp[(size_t)i * NC] = v;
        }
    }
}

// ---------------------------------------------------------------------------
// BatchNorm: column sums / sums-of-squares (coalesced), finalize, apply.
// ---------------------------------------------------------------------------
__global__ void zero_f32_kernel(float* __restrict__ p, int n) {
    int i = blockIdx.x * blockDim.x + threadIdx.x;
    if (i < n) p[i] = 0.f;
}

__global__ void bn_stats_kernel(const float* __restrict__ z,
                                float* __restrict__ csum,
                                float* __restrict__ csq, int n) {
    const int ROWS = 128;
    int c = threadIdx.x;                       // 256 threads = 256 columns
    int row0 = blockIdx.x * ROWS;
    float s = 0.f, s2 = 0.f;
    for (int r = 0; r < ROWS; ++r) {
        int row = row0 + r;
        if (row < n) {
            float v = z[(size_t)row * HID + c];
            s += v;
            s2 += v * v;
        }
    }
    atomicAdd(&csum[c], s);
    atomicAdd(&csq[c], s2);
}

__global__ void bn_finalize_kernel(const float* __restrict__ csum,
                                   const float* __restrict__ csq,
                                   float* __restrict__ mu,
                                   float* __restrict__ rstd, int n) {
    int c = threadIdx.x;
    float inv_n = 1.f / (float)n;
    float m = csum[c] * inv_n;
    float var = csq[c] * inv_n - m * m;
    mu[c] = m;
    rstd[c] = rsqrtf(var + BN_EPS);
}

// h = relu((z - mu) * rstd * g + b) + h   (residual, in place on h)
__global__ void bn_apply_kernel(const float* __restrict__ z,
                                const float* __restrict__ mu,
                                const float* __restrict__ rstd,
                                const float* __restrict__ g,
                                const float* __restrict__ b,
                                float* __restrict__ h) {
    int c = threadIdx.x;                       // 256 threads, HID=256 cols
    size_t i = (size_t)blockIdx.x * HID + c;
    float v = (z[i] - mu[c]) * rstd[c] * g[c] + b[c];
    v = v > 0.f ? v : 0.f;
    h[i] += v;
}

// ---------------------------------------------------------------------------
// Per-graph pooling: sum / count / max with atomics.
// ---------------------------------------------------------------------------
__device__ inline void atomicMaxF(float* addr, float v) {
    if (v >= 0.f)
        atomicMax((int*)addr, __float_as_int(v));
    else
        atomicMin((unsigned int*)addr, (unsigned int)__float_as_int(v));
}

__global__ void pool_init_kernel(float* __restrict__ psum,
                                 float* __restrict__ pmax,
                                 float* __restrict__ pcnt) {
    int i = blockIdx.x * blockDim.x + threadIdx.x;   // NGRAPH*HID threads
    psum[i] = 0.f;
    pmax[i] = -3.402823466e38f;
    if (i < NGRAPH) pcnt[i] = 0.f;
}

__global__ void pool_kernel(const float* __restrict__ h,
                            const int* __restrict__ batch,
                            float* __restrict__ psum,
                            float* __restrict__ pmax,
                            float* __restrict__ pcnt) {
    __shared__ int gidx;
    int node = blockIdx.x;
    int c = threadIdx.x;
    if (c == 0) gidx = batch[node];
    __syncthreads();
    int g = gidx;
    float v = h[(size_t)node * HID + c];
    atomicAdd(&psum[(size_t)g * HID + c], v);
    atomicMaxF(&pmax[(size_t)g * HID + c], v);
    if (c == 0) atomicAdd(&pcnt[g], 1.f);
}

// g = concat([mean, sum, max])  -> [G, 768]
__global__ void build_g_kernel(const float* __restrict__ psum,
                               const float* __restrict__ pmax,
                               const float* __restrict__ pcnt,
                               float* __restrict__ gfeat) {
    int gi = blockIdx.x;
    int c = threadIdx.x;
    float cn = pcnt[gi];
    cn = cn > 1.f ? cn : 1.f;
    float s = psum[(size_t)gi * HID + c];
    float m = pmax[(size_t)gi * HID + c];
    gfeat[(size_t)gi * 768 + c]       = s / cn;
    gfeat[(size_t)gi * 768 + 256 + c] = s;
    gfeat[(size_t)gi * 768 + 512 + c] = m;
}

// out = g2 @ head_W3 + head_b3   ([G,128] x [128,1])
__global__ void head_final_kernel(const float* __restrict__ g2,
                                  const float* __restrict__ W3,
                                  const float* __restrict__ b3,
                                  float* __restrict__ out) {
    int gi = blockIdx.x * blockDim.x + threadIdx.x;
    if (gi >= NGRAPH) return;
    float s = b3[0];
#pragma unroll 8
    for (int k = 0; k < 128; ++k)
        s += g2[(size_t)gi * 128 + k] * W3[k];
    out[gi] = s;
}

// ---------------------------------------------------------------------------
static inline void launch_gemm(const float* A, const float* B, const float* bias,
                               float* C, int M, int Nc, int K, int relu,
                               hipStream_t s) {
    size_t lds = (size_t)16 * K * sizeof(float);
    if (K == 256 && Nc == 256)
        gemm_wmma_tiled<256, 256><<<M / 16, 256, lds, s>>>(A, B, bias, C, relu);
    else if (K == 768 && Nc == 256)
        gemm_wmma_tiled<768, 256><<<M / 16, 256, lds, s>>>(A, B, bias, C, relu);
    else if (K == 256 && Nc == 128)
        gemm_wmma_tiled<256, 128><<<M / 16, 256, lds, s>>>(A, B, bias, C, relu);
}

extern "C" void kernel_launch(void* const* d_in, const int* in_sizes, int n_in,
                              void* d_out, int out_size, void* d_ws, size_t ws_size,
                              hipStream_t stream) {
    const float* x      = (const float*)d_in[0];
    const int*   ei     = (const int*)d_in[1];
    const float* ea     = (const float*)d_in[2];
    const int*   batch  = (const int*)d_in[3];
    const float* enc_W  = (const float*)d_in[4];
    const float* enc_b  = (const float*)d_in[5];
    const float* edge_W = (const float*)d_in[6];   // [L,5,256]
    const float* edge_b = (const float*)d_in[7];   // [L,256]
    const float* mlp_W1 = (const float*)d_in[8];   // [L,256,256]
    const float* mlp_b1 = (const float*)d_in[9];
    const float* mlp_W2 = (const float*)d_in[10];
    const float* mlp_b2 = (const float*)d_in[11];
    const float* bn_g   = (const float*)d_in[12];
    const float* bn_b   = (const float*)d_in[13];
    const float* hW1    = (const float*)d_in[14];  // [768,256]
    const float* hb1    = (const float*)d_in[15];
    const float* hW2    = (const float*)d_in[16];  // [256,128]
    const float* hb2    = (const float*)d_in[17];
    const float* hW3    = (const float*)d_in[18];  // [128,1]
    const float* hb3    = (const float*)d_in[19];
    float* out = (float*)d_out;

    const size_t NHB = (size_t)N_NODES * HID * sizeof(float);   // 51.2 MB
    char* ws = (char*)d_ws;
    float* h      = (float*)(ws);
    float* t1     = (float*)(ws + NHB);            // aggr, then z (post-MLP)
    float* z1     = (float*)(ws + 2 * NHB);        // hidden of MLP
    float* csum   = (float*)(ws + 3 * NHB);        // [256]
    float* csq    = csum + HID;                    // [256]
    float* mu     = csum + 2 * HID;                // [256]
    float* rstd   = csum + 3 * HID;                // [256]
    float* psum   = (float*)(ws + 3 * NHB + 4096);           // [G,256]
    float* pmax   = psum + (size_t)NGRAPH * HID;             // [G,256]
    float* pcnt   = pmax + (size_t)NGRAPH * HID;             // [G]
    float* gfeat  = pcnt + NGRAPH;                           // [G,768]
    float* g1     = gfeat + (size_t)NGRAPH * 768;            // [G,256]
    float* g2     = g1 + (size_t)NGRAPH * HID;               // [G,128]

    // 1) node encoder
    encoder_kernel<<<N_NODES, 256, 0, stream>>>(x, enc_W, enc_b, h);

    // 2) GINE layers
    for (int l = 0; l < NLAYERS; ++l) {
        // aggr := h  (folds z = h + sum(msg))
        hipMemcpyAsync(t1, h, NHB, hipMemcpyDeviceToDevice, stream);
        edge_aggr_kernel<<<N_EDGES / 8, 256, 0, stream>>>(
            ea, ei, h, edge_W + (size_t)l * 5 * HID, edge_b + (size_t)l * HID,
            t1, N_EDGES);
        // z1 = relu(t1 @ W1 + b1)
        launch_gemm(t1, mlp_W1 + (size_t)l * HID * HID,
                    mlp_b1 + (size_t)l * HID, z1, N_NODES, HID, HID, 1, stream);
        // z (into t1) = z1 @ W2 + b2
        launch_gemm(z1, mlp_W2 + (size_t)l * HID * HID,
                    mlp_b2 + (size_t)l * HID, t1, N_NODES, HID, HID, 0, stream);
        // batchnorm stats + apply + relu + residual into h
        zero_f32_kernel<<<2, 256, 0, stream>>>(csum, 2 * HID);
        bn_stats_kernel<<<(N_NODES + 127) / 128, 256, 0, stream>>>(
            t1, csum, csq, N_NODES);
        bn_finalize_kernel<<<1, 256, 0, stream>>>(csum, csq, mu, rstd, N_NODES);
        bn_apply_kernel<<<N_NODES, 256, 0, stream>>>(
            t1, mu, rstd, bn_g + (size_t)l * HID, bn_b + (size_t)l * HID, h);
    }

    // 3) per-graph pooling (mean / sum / max)
    pool_init_kernel<<<NGRAPH * HID / 256, 256, 0, stream>>>(psum, pmax, pcnt);
    pool_kernel<<<N_NODES, 256, 0, stream>>>(h, batch, psum, pmax, pcnt);
    build_g_kernel<<<NGRAPH, 256, 0, stream>>>(psum, pmax, pcnt, gfeat);

    // 4) head MLP
    launch_gemm(gfeat, hW1, hb1, g1, NGRAPH, HID, 3 * HID, 1, stream);   // [1024,768]x[768,256]
    launch_gemm(g1, hW2, hb2, g2, NGRAPH, 128, HID, 1, stream);          // [1024,256]x[256,128]
    head_final_kernel<<<NGRAPH / 256, 256, 0, stream>>>(g2, hW3, hb3, out);
}